// Simple_MLP_47330539602473
// MI455X (gfx1250) — compile-verified
//
#include <hip/hip_runtime.h>
#include <hip/hip_bf16.h>

// ---------------------------------------------------------------------------
// Simple_MLP on MI455X (gfx1250, wave32)
//
// out[b] = sum_p sum_h softplus( sum_d x[b, perm[p,d]] * w1[h,d] + b1[h] ) * w2[h]
//   BATCH=4096, NPERM=720, DIM=12, HIDDEN=128
//
// * One block (256 thr = 8 waves) per batch row. 720 perms = 45 tiles of 16.
// * Dense contraction (K=12 = 3 x K4) via V_WMMA_F32_16X16X4_F32 against the
//   8 N-tiles of HIDDEN=128. w1/b1/w2 fragments preloaded (loop-invariant);
//   per-tile A fragments gathered from the wave-resident 12-float x row via
//   ds_bpermute shuffles.
// * softplus with raw v_exp_f32/v_log_f32 (base-2 identity):
//     sp = max(h,0) + ln2 * log2(1 + exp2(-|h| * log2e))
//   exp2 arg <= 0 -> result in (0,1]; log2 arg in (1,2]: no denorm/inf edge
//   cases, so bare hardware trans ops are safe. Computation is STAGED across
//   the 8 independent tile elements (exp batch -> add batch -> log batch ->
//   fma batch) so TRANS hazard slots fill with real work, not v_nop, and
//   accumulation uses 4 rotating registers to break the fmac serial chain.
// * Flat shfl_xor wave reduce + 8-slot LDS block reduce; no atomics.
// ---------------------------------------------------------------------------

typedef __attribute__((ext_vector_type(2))) float v2f;
typedef __attribute__((ext_vector_type(8))) float v8f;
typedef __attribute__((ext_vector_type(2))) int   v2i;

namespace {
constexpr int DIM    = 12;
constexpr int HIDDEN = 128;
constexpr int NPERM  = 720;
constexpr int PTILES = NPERM / 16;   // 45
constexpr int NTILES = HIDDEN / 16;  // 8
constexpr int WAVES  = 8;            // 256 threads / 32
}

__global__ __launch_bounds__(256, 2)
void simple_mlp_perm_wmma(const float* __restrict__ x,
                          const float* __restrict__ w1,
                          const float* __restrict__ b1,
                          const float* __restrict__ w2,
                          const int*   __restrict__ perm,
                          float*       __restrict__ out)
{
    const int b    = blockIdx.x;
    const int lane = threadIdx.x & 31;
    const int wave = threadIdx.x >> 5;
    const int half = lane >> 4;      // 0: lanes 0-15, 1: lanes 16-31
    const int l16  = lane & 15;

    constexpr float LOG2E = 1.44269504088896340736f;
    constexpr float LN2   = 0.69314718055994530942f;

    // Broadcast this batch row's 12 x-values across the wave.
    const float xlane = x[b * DIM + (lane < DIM ? lane : 0)];

    // ---- Preload all loop-invariant fragments into registers ----
    // B layout for 16x16x4 f32 (N across lanes): lane L, n = L%16,
    //   k-pair base = 2*(L/16); step s covers K = 4s+kb, 4s+kb+1.
    v2f   bfrag[NTILES][3];
    float bias[NTILES];
    float wo[NTILES];
#pragma unroll
    for (int nt = 0; nt < NTILES; ++nt) {
        const int n = nt * 16 + l16;
#pragma unroll
        for (int s = 0; s < 3; ++s) {
            const int k0 = 4 * s + 2 * half;          // even -> 8B aligned
            bfrag[nt][s] = *(const v2f*)(w1 + n * DIM + k0);
        }
        bias[nt] = b1[n];
        wo[nt]   = w2[n];
    }

    float acc0 = 0.0f, acc1 = 0.0f, acc2 = 0.0f, acc3 = 0.0f;

    for (int t = wave; t < PTILES; t += WAVES) {
        const int p = t * 16 + l16;                   // perm row (half-lane)

        // Gather the 3 A fragments (K=12 -> 3 steps of K4) via shuffles.
        v2f afrag[3];
#pragma unroll
        for (int s = 0; s < 3; ++s) {
            const int k0 = 4 * s + 2 * half;          // even -> 8B aligned
            const v2i pi = *(const v2i*)(perm + p * DIM + k0);
            afrag[s].x = __shfl(xlane, pi.x, 32);
            afrag[s].y = __shfl(xlane, pi.y, 32);
        }

#pragma unroll
        for (int nt = 0; nt < NTILES; ++nt) {
            v8f c = {};                               // 16x16 f32 tile
#pragma unroll
            for (int s = 0; s < 3; ++s) {
                c = __builtin_amdgcn_wmma_f32_16x16x4_f32(
                        false, afrag[s], false, bfrag[nt][s],
                        (short)0, c, false, false);
            }

            const float bi = bias[nt];
            const float wn = wo[nt];

            // ---- Staged softplus over the 8 independent elements ----
            float hmax[8], e[8], a1[8], l[8];
#pragma unroll
            for (int i = 0; i < 8; ++i) {
                const float h = c[i] + bi;
                hmax[i] = fmaxf(h, 0.0f);
                e[i] = __builtin_amdgcn_exp2f(-fabsf(h) * LOG2E);
            }
#pragma unroll
            for (int i = 0; i < 8; ++i)
                a1[i] = 1.0f + e[i];
#pragma unroll
            for (int i = 0; i < 8; ++i)
                l[i] = __builtin_amdgcn_logf(a1[i]);

            acc0 = fmaf(fmaf(LN2, l[0], hmax[0]), wn, acc0);
            acc1 = fmaf(fmaf(LN2, l[1], hmax[1]), wn, acc1);
            acc2 = fmaf(fmaf(LN2, l[2], hmax[2]), wn, acc2);
            acc3 = fmaf(fmaf(LN2, l[3], hmax[3]), wn, acc3);
            acc0 = fmaf(fmaf(LN2, l[4], hmax[4]), wn, acc0);
            acc1 = fmaf(fmaf(LN2, l[5], hmax[5]), wn, acc1);
            acc2 = fmaf(fmaf(LN2, l[6], hmax[6]), wn, acc2);
            acc3 = fmaf(fmaf(LN2, l[7], hmax[7]), wn, acc3);
        }
    }

    float lane_acc = (acc0 + acc1) + (acc2 + acc3);

    // Wave32 reduction of the flat partial.
#pragma unroll
    for (int off = 16; off >= 1; off >>= 1)
        lane_acc += __shfl_xor(lane_acc, off, 32);

    __shared__ float red[WAVES];
    if (lane == 0) red[wave] = lane_acc;
    __syncthreads();

    if (threadIdx.x == 0) {
        float tot = 0.0f;
#pragma unroll
        for (int i = 0; i < WAVES; ++i) tot += red[i];
        out[b] = tot;
    }
}

extern "C" void kernel_launch(void* const* d_in, const int* in_sizes, int n_in,
                              void* d_out, int out_size, void* d_ws, size_t ws_size,
                              hipStream_t stream)
{
    (void)in_sizes; (void)n_in; (void)d_ws; (void)ws_size;

    const float* x    = (const float*)d_in[0];  // (4096, 12)
    const float* w1   = (const float*)d_in[1];  // (128, 12) row-major
    const float* b1   = (const float*)d_in[2];  // (128,)
    const float* w2   = (const float*)d_in[3];  // (128,)
    const int*   perm = (const int*)  d_in[4];  // (720, 12)
    float*       out  = (float*)d_out;          // (4096,)

    simple_mlp_perm_wmma<<<out_size, 256, 0, stream>>>(x, w1, b1, w2, perm, out);
}